// IDWTLayer_69750268887082
// MI455X (gfx1250) — compile-verified
//
#include <hip/hip_runtime.h>
#include <stdint.h>
#include <stddef.h>

typedef __attribute__((ext_vector_type(2))) float v2f;
typedef __attribute__((ext_vector_type(8))) float v8f;
typedef __attribute__((ext_vector_type(4))) unsigned int v4u;
typedef __attribute__((ext_vector_type(4))) int v4i;
typedef __attribute__((ext_vector_type(8))) int v8i;

#define BATCH 16
#define LIN   32768
#define CIN   64
#define COUT  32
#define LOUT  65536
#define TROWS 36   // LDS tile rows: m in [Mb-1, Mb+34]
#define WGSZ  256

__global__ __launch_bounds__(WGSZ)
void idwt_wmma_kernel(const float* __restrict__ x,
                      const float* __restrict__ rec_lo,
                      const float* __restrict__ rec_hi,
                      float* __restrict__ out)
{
    __shared__ float tile[TROWS][CIN];   // 36*64*4 = 9216 B

    const int tb  = blockIdx.x;          // 0..1023 : block of 64 output rows
    const int b   = blockIdx.y;          // 0..15   : batch
    const int tid = threadIdx.x;

    const int Mb      = tb * 32;         // first m handled by this WG
    const int m_start = Mb - 1;          // first tile row (may be -1)
    const int head    = (m_start < 0) ? 1 : 0;
    const int m0      = m_start + head;  // first DMA'd row
    const int nrows   = TROWS - head;    // rows to DMA
    const unsigned rem_rows = (unsigned)(LIN - m0); // rows left in tensor; tail OOB reads -> 0

    // conv zero-padding at the sequence head: zero tile row 0, DMA shifted by one row
    if (head && tid < CIN) tile[0][tid] = 0.0f;

    // ---- Tensor Data Mover: DMA (nrows x 64) fp32 tile global -> LDS ----
    if (tid < 32) {  // wave 0 only (uniform branch); TDM ignores EXEC
        const uint64_t gaddr = (uint64_t)(uintptr_t)(x + ((size_t)b * LIN + (size_t)m0) * CIN);
        const uint32_t laddr = (uint32_t)(uintptr_t)(&tile[head][0]);

        v4u g0;
        g0.x = 1u;                                        // count=1 (valid), user mode
        g0.y = laddr;                                     // lds_addr (bytes)
        g0.z = (uint32_t)gaddr;                           // global_addr[31:0]
        g0.w = ((uint32_t)(gaddr >> 32) & 0x01FFFFFFu)    // global_addr[56:32]
               | (2u << 30);                              // type=2 ("image")

        v8i g1;
        g1[0] = (int)(2u << 16);                          // wg_mask=0, data_size=2 (4B)
        g1[1] = (int)(64u << 16);                         // tensor_dim0[15:0]=64 @ bits63:48
        g1[2] = (int)((rem_rows & 0xFFFFu) << 16);        // dim0[31:16]=0 | tensor_dim1 lo
        g1[3] = (int)(((rem_rows >> 16) & 0xFFFFu)        // tensor_dim1 hi
               | (64u << 16));                            // tile_dim0 = 64
        g1[4] = (int)(nrows & 0xFFFF);                    // tile_dim1 = nrows, tile_dim2=0
        g1[5] = 64;                                       // tensor_dim0_stride = 64 elems
        g1[6] = 0;                                        // stride0 hi | stride1 lo
        g1[7] = 0;                                        // stride1 hi (unused for 2D)

        v4i gz4 = {0, 0, 0, 0};                           // groups 2/3 unused (2D tile)
        v8i gz8 = {0, 0, 0, 0, 0, 0, 0, 0};
        __builtin_amdgcn_tensor_load_to_lds(g0, g1, gz4, gz4, gz8, 0);
        __builtin_amdgcn_s_wait_tensorcnt(0);
    }
    __syncthreads();

    // ---- per-wave 16x16 output tile via V_WMMA_F32_16X16X4_F32 ----
    const int wave  = tid >> 5;          // 0..7
    const int lane  = tid & 31;
    const int tblk  = wave >> 1;         // 0..3 : which 16 output rows
    const int chalf = wave & 1;          // 0/1  : which 16 channels
    const int n     = lane & 15;         // A-matrix row M and B/C/D column N
    const int h     = lane >> 4;         // lane-half selector
    const int j0    = tblk * 8;          // tile-row base (m = Mb-1+j0 maps to j=0)

    // Banded coefficient matrix A[16][12]:
    // output row M (parity p) draws from tile cols j = (M>>1)+i, i=0..3,
    // with tap f[2*i + 1 - p].  Fetch straight from global with a clamped
    // per-lane offset (coalesces into one cacheline; hot in L0 after WG 0)
    // -> no register-array dynamic indexing, no cndmask select chains.
    const int p  = n & 1;
    const int rh = n >> 1;
    auto coef = [&](const float* __restrict__ f, int j) -> float {
        const int  i     = j - rh;
        const bool valid = (unsigned)i <= 3u;
        int idx = 2 * i + 1 - p;
        idx = idx < 0 ? 0 : (idx > 7 ? 7 : idx);   // keep load in-range (v_med3)
        const float v = f[idx];
        return valid ? v : 0.0f;
    };

    const int ca = chalf * 16 + n;       // approx channel; detail = ca + 32

    v8f acc = {0.f, 0.f, 0.f, 0.f, 0.f, 0.f, 0.f, 0.f};
#pragma unroll
    for (int kc = 0; kc < 3; ++kc) {
        const int ja = kc * 4 + 2 * h;   // K held in component .x ; .y holds K+1

        v2f a_lo; a_lo.x = coef(rec_lo, ja); a_lo.y = coef(rec_lo, ja + 1);
        v2f a_hi; a_hi.x = coef(rec_hi, ja); a_hi.y = coef(rec_hi, ja + 1);

        const float* r0 = &tile[j0 + ja][ca];
        const float* r1 = &tile[j0 + ja + 1][ca];
        v2f bA; bA.x = r0[0];  bA.y = r1[0];   // ds_load_2addr pairs
        v2f bD; bD.x = r0[32]; bD.y = r1[32];

        acc = __builtin_amdgcn_wmma_f32_16x16x4_f32(false, a_lo, false, bA,
                                                    (short)0, acc, false, false);
        acc = __builtin_amdgcn_wmma_f32_16x16x4_f32(false, a_hi, false, bD,
                                                    (short)0, acc, false, false);
    }

    // ---- store: D VGPR v holds output row t0 + v + 8*h, channel ca ----
    const int t0 = tb * 64 + tblk * 16;
    float* outp = out + ((size_t)b * LOUT + (size_t)(t0 + 8 * h)) * COUT + ca;
#pragma unroll
    for (int v = 0; v < 8; ++v) {
        outp[(size_t)v * COUT] = acc[v];
    }
}

extern "C" void kernel_launch(void* const* d_in, const int* in_sizes, int n_in,
                              void* d_out, int out_size, void* d_ws, size_t ws_size,
                              hipStream_t stream) {
    const float* x      = (const float*)d_in[0];
    const float* rec_lo = (const float*)d_in[1];
    const float* rec_hi = (const float*)d_in[2];
    float* out = (float*)d_out;
    (void)in_sizes; (void)n_in; (void)out_size; (void)d_ws; (void)ws_size;

    dim3 grid(LOUT / 64, BATCH);   // 1024 x 16 workgroups
    dim3 block(WGSZ);
    idwt_wmma_kernel<<<grid, block, 0, stream>>>(x, rec_lo, rec_hi, out);
}